// Pooler_46866683134507
// MI455X (gfx1250) — compile-verified
//
#include <hip/hip_runtime.h>
#include <math.h>

// Problem constants (fixed by the reference)
#define B_   2
#define C_   256
#define H_   256
#define W_   256
#define N_   512
#define R_   (B_ * N_)        // 1024 rois
#define OUT_H 7
#define OUT_W 7
#define NBIN  (OUT_H * OUT_W) // 49
#define SPATIAL_SCALE 0.25f

#define AS1 __attribute__((address_space(1)))
#define AS3 __attribute__((address_space(3)))

#if __has_builtin(__builtin_amdgcn_global_load_async_to_lds_b32) && \
    __has_builtin(__builtin_amdgcn_s_wait_asynccnt)
#define HAVE_ASYNC 1
#else
#define HAVE_ASYNC 0
#endif

// ---------------------------------------------------------------------------
// Pass 1: NCHW -> NHWC layout transform (makes the per-bin channel gather
// contiguous: 1KB per corner instead of 256 cachelines).
// grid = (64, 256, 2) : blockIdx.x = ctile(0..7) + 8*xtile(0..7), y, b
// block = (32, 8)
// ---------------------------------------------------------------------------
__global__ void __launch_bounds__(256)
transpose_nchw_to_nhwc(const float* __restrict__ x, float* __restrict__ ws) {
    __shared__ float tile[32][33];   // +1 pad: conflict-free transposed reads
    const int ctile = blockIdx.x & 7;
    const int xtile = blockIdx.x >> 3;
    const int y  = blockIdx.y;
    const int b  = blockIdx.z;
    const int tx = threadIdx.x;      // 0..31
    const int ty = threadIdx.y;      // 0..7
    const int c0 = ctile * 32;
    const int x0 = xtile * 32;

#pragma unroll
    for (int k = 0; k < 4; ++k) {
        const int cl = ty + 8 * k;               // channel within tile
        const int c  = c0 + cl;
        const size_t gidx = ((size_t)((b * C_ + c) * H_ + y)) * W_ + (x0 + tx);
#if HAVE_ASYNC
        // CDNA5 async copy engine: global -> LDS without VGPR round trip,
        // tracked by ASYNCcnt. Builtin prototype: (global int*, local int*,
        // imm offset, imm cpol).
        __builtin_amdgcn_global_load_async_to_lds_b32(
            (AS1 int*)(x + gidx), (AS3 int*)&tile[cl][tx], 0, 0);
#else
        tile[cl][tx] = x[gidx];
#endif
    }
#if HAVE_ASYNC
    __builtin_amdgcn_s_wait_asynccnt(0);
#endif
    __syncthreads();

#pragma unroll
    for (int k = 0; k < 4; ++k) {
        const int xl = ty + 8 * k;               // x within tile
        const int xg = x0 + xl;
        // ws[((b*H + y)*W + x)*C + c], coalesced along c (= tx)
        const size_t widx = ((size_t)((b * H_ + y) * W_ + xg)) * C_ + (c0 + tx);
        ws[widx] = tile[tx][xl];
    }
}

// ---------------------------------------------------------------------------
// Per-bin geometry, exactly mirroring the JAX reference (rint = round-to-
// nearest-even, floor/ceil of bin center, validity-masked bilinear weights).
// Threads 0..48 each compute one bin.
// ---------------------------------------------------------------------------
__device__ __forceinline__ void compute_bin_geometry(
    const float* __restrict__ boxes, int roi, int tid,
    int4* s_idx, float4* s_w) {
    if (tid >= NBIN) return;
    const float* bx = boxes + (size_t)roi * 5;
    const float cx  = bx[0] * SPATIAL_SCALE;
    const float cy  = bx[1] * SPATIAL_SCALE;
    const float w   = bx[2] * SPATIAL_SCALE;
    const float h   = bx[3] * SPATIAL_SCALE;
    const float ang = bx[4] * 0.017453292519943295f; // pi/180
    const float Sx = w * (1.0f / OUT_W);
    const float Sy = h * (1.0f / OUT_H);
    const float ca = cosf(ang), sa = sinf(ang);
    const float dx = -0.5f * OUT_W, dy = -0.5f * OUT_H;
    const float M00 = ca * Sx,  M01 = sa * Sy;
    const float M02 = ca * Sx * dx + sa * Sy * dy + cx;
    const float M10 = -sa * Sx, M11 = ca * Sy;
    const float M12 = -sa * Sx * dx + ca * Sy * dy + cy;

    const int ph = tid / OUT_W;
    const int pw = tid % OUT_W;
    float xmn =  3.0e38f, xmx = -3.0e38f, ymn =  3.0e38f, ymx = -3.0e38f;
#pragma unroll
    for (int oy = 0; oy < 2; ++oy) {
#pragma unroll
        for (int ox = 0; ox < 2; ++ox) {
            const float px = (float)(pw + ox);
            const float py = (float)(ph + oy);
            const float X = M00 * px + M01 * py + M02;
            const float Y = M10 * px + M11 * py + M12;
            xmn = fminf(xmn, X); xmx = fmaxf(xmx, X);
            ymn = fminf(ymn, Y); ymx = fmaxf(ymx, Y);
        }
    }
    const float leftMost  = fmaxf(rintf(xmn), 0.0f);
    const float rightMost = fminf(rintf(xmx), (float)(W_ - 1));
    const float topMost   = fmaxf(rintf(ymn), 0.0f);
    const float botMost   = fminf(rintf(ymx), (float)(H_ - 1));
    const float bcx = 0.5f * (leftMost + rightMost);
    const float bcy = 0.5f * (topMost + botMost);
    const float flx = floorf(bcx), fly = floorf(bcy);
    int xl = (int)flx,        xr = (int)ceilf(bcx);
    int yt = (int)fly,        yb = (int)ceilf(bcy);
    const float rx = bcx - flx, ry = bcy - fly;
    float wlt = (1.0f - rx) * (1.0f - ry);
    float wrt = rx * (1.0f - ry);
    float wrb = rx * ry;
    float wlb = (1.0f - rx) * ry;
    const bool vl = (xl >= 0) && (xl < W_);
    const bool vr = (xr >= 0) && (xr < W_);
    const bool vt = (yt >= 0) && (yt < H_);
    const bool vb = (yb >= 0) && (yb < H_);
    wlt = (vt && vl) ? wlt : 0.0f;
    wrt = (vt && vr) ? wrt : 0.0f;
    wrb = (vb && vr) ? wrb : 0.0f;
    wlb = (vb && vl) ? wlb : 0.0f;
    xl = min(max(xl, 0), W_ - 1); xr = min(max(xr, 0), W_ - 1);
    yt = min(max(yt, 0), H_ - 1); yb = min(max(yb, 0), H_ - 1);
    s_idx[tid] = make_int4(xl, xr, yt, yb);
    s_w[tid]   = make_float4(wlt, wrt, wrb, wlb);
}

// ---------------------------------------------------------------------------
// Pass 2 (fast path): gather from NHWC workspace.
// One block (256 thr = 8 wave32) per ROI; thread == channel.
// Output staged in LDS so the global store is one contiguous float4 burst.
// ---------------------------------------------------------------------------
__global__ void __launch_bounds__(256)
rroi_gather_nhwc(const float* __restrict__ ws, const float* __restrict__ boxes,
                 float* __restrict__ out) {
    __shared__ int4   s_idx[NBIN];
    __shared__ float4 s_w[NBIN];
    __shared__ __align__(16) float s_out[C_ * NBIN];   // 49 KB staging

    const int roi = blockIdx.x;
    const int tid = threadIdx.x;
    const int b   = roi / N_;

    compute_bin_geometry(boxes, roi, tid, s_idx, s_w);
    __syncthreads();

    const int c = tid;  // 256 threads == 256 channels
    const float* fb = ws + (size_t)b * H_ * W_ * C_ + c;
    __builtin_prefetch(fb, 0, 0);   // -> global_prefetch_b8

    for (int bin = 0; bin < NBIN; ++bin) {
        const int4   id = s_idx[bin];
        const float4 wt = s_w[bin];
        const float vlt = fb[((size_t)id.z * W_ + id.x) * C_]; // (yt, xl)
        const float vrt = fb[((size_t)id.z * W_ + id.y) * C_]; // (yt, xr)
        const float vrb = fb[((size_t)id.w * W_ + id.y) * C_]; // (yb, xr)
        const float vlb = fb[((size_t)id.w * W_ + id.x) * C_]; // (yb, xl)
        const float acc = vlt * wt.x + vrt * wt.y + vrb * wt.z + vlb * wt.w;
        s_out[c * NBIN + bin] = fmaxf(acc, 0.0f);   // stride-49: bank-free
    }
    __syncthreads();

    const float4* so4 = (const float4*)s_out;
    float4* o4 = (float4*)(out + (size_t)roi * (C_ * NBIN));
    for (int i = tid; i < (C_ * NBIN) / 4; i += 256) o4[i] = so4[i];
}

// ---------------------------------------------------------------------------
// Fallback: direct NCHW gather (only if workspace is too small).
// ---------------------------------------------------------------------------
__global__ void __launch_bounds__(256)
rroi_gather_nchw(const float* __restrict__ x, const float* __restrict__ boxes,
                 float* __restrict__ out) {
    __shared__ int4   s_idx[NBIN];
    __shared__ float4 s_w[NBIN];
    __shared__ __align__(16) float s_out[C_ * NBIN];

    const int roi = blockIdx.x;
    const int tid = threadIdx.x;
    const int b   = roi / N_;

    compute_bin_geometry(boxes, roi, tid, s_idx, s_w);
    __syncthreads();

    const int c = tid;
    const float* fb = x + ((size_t)(b * C_ + c)) * H_ * W_;
    __builtin_prefetch(fb, 0, 0);

    for (int bin = 0; bin < NBIN; ++bin) {
        const int4   id = s_idx[bin];
        const float4 wt = s_w[bin];
        const float vlt = fb[(size_t)id.z * W_ + id.x];
        const float vrt = fb[(size_t)id.z * W_ + id.y];
        const float vrb = fb[(size_t)id.w * W_ + id.y];
        const float vlb = fb[(size_t)id.w * W_ + id.x];
        const float acc = vlt * wt.x + vrt * wt.y + vrb * wt.z + vlb * wt.w;
        s_out[c * NBIN + bin] = fmaxf(acc, 0.0f);
    }
    __syncthreads();

    const float4* so4 = (const float4*)s_out;
    float4* o4 = (float4*)(out + (size_t)roi * (C_ * NBIN));
    for (int i = tid; i < (C_ * NBIN) / 4; i += 256) o4[i] = so4[i];
}

// ---------------------------------------------------------------------------
extern "C" void kernel_launch(void* const* d_in, const int* in_sizes, int n_in,
                              void* d_out, int out_size, void* d_ws, size_t ws_size,
                              hipStream_t stream) {
    (void)in_sizes; (void)n_in; (void)out_size;
    const float* x     = (const float*)d_in[0];   // (2,256,256,256) f32
    const float* boxes = (const float*)d_in[1];   // (2,512,5) f32
    float* out = (float*)d_out;                   // (1024,256,7,7) f32

    const size_t need = (size_t)B_ * H_ * W_ * C_ * sizeof(float); // 128 MB
    if (d_ws != nullptr && ws_size >= need) {
        float* wsf = (float*)d_ws;
        // NCHW -> NHWC
        transpose_nchw_to_nhwc<<<dim3(64, 256, 2), dim3(32, 8, 1), 0, stream>>>(x, wsf);
        // Coalesced gather
        rroi_gather_nhwc<<<dim3(R_), dim3(256), 0, stream>>>(wsf, boxes, out);
    } else {
        rroi_gather_nchw<<<dim3(R_), dim3(256), 0, stream>>>(x, boxes, out);
    }
}